// FrequencyBranch_43293270344063
// MI455X (gfx1250) — compile-verified
//
#include <hip/hip_runtime.h>
#include <math.h>

// Problem constants (from reference): B=4, C=3, W=256, H=129, N=10
#define BD 4
#define CD 3
#define WD 256
#define HD 129
#define NP 10
#define ROWS (BD * CD)          // 12
#define WH (WD * HD)            // 33024 (divisible by 4)

typedef float v4f __attribute__((ext_vector_type(4)));

// ---------------------------------------------------------------------------
// Kernel 1: 24 independent sequential greedy interval-NMS problems (x band and
// y band for each of 12 (b,c) rows), fully in registers. Writes quantized
// integer bounds (post ind-zeroing, floor, clip) to workspace and the 120
// p = sqrt(px_*py_ + 1e-10) outputs.
// ws layout per row (40 ints): [x1[10] | x2[10] | y1[10] | y2[10]]
// ---------------------------------------------------------------------------
__global__ __launch_bounds__(32) void nms_quant_kernel(
    const float* __restrict__ px,  const float* __restrict__ xc1, const float* __restrict__ xc2,
    const float* __restrict__ py,  const float* __restrict__ yc1, const float* __restrict__ yc2,
    int* __restrict__ wsq, float* __restrict__ pout) {
  __shared__ float spx[ROWS * NP];
  __shared__ float spy[ROWS * NP];
  const int t = threadIdx.x;

  if (t < 2 * ROWS) {
    const int band = t / ROWS;       // 0 = x, 1 = y
    const int row  = t % ROWS;       // b*C + c
    const float* P  = band ? py  : px;
    const float* C1 = band ? yc1 : xc1;
    const float* C2 = band ? yc2 : xc2;
    const float dim = band ? (float)HD : (float)WD;

    float pv[NP], a1[NP], a2[NP];
    int   ind[NP];
#pragma unroll
    for (int i = 0; i < NP; ++i) {
      pv[i] = P[row * NP + i];
      a1[i] = C1[row * NP + i];
      a2[i] = C2[row * NP + i];
      ind[i] = (fmaxf(a2[i] - a1[i], 0.0f) > 0.0f) ? 1 : 0;
    }

    // Greedy NMS in stable descending-score order (matches jnp.argsort(-p)).
    unsigned used = 0u;
    for (int k = 0; k < NP; ++k) {
      int best = 0;
      float bp = -1e30f;
#pragma unroll
      for (int i = 0; i < NP; ++i) {
        if (!((used >> i) & 1u) && pv[i] > bp) { bp = pv[i]; best = i; }
      }
      used |= 1u << best;
      if (ind[best]) {
        const float b1 = a1[best], b2 = a2[best];
#pragma unroll
        for (int j = 0; j < NP; ++j) {
          const float lo  = fmaxf(a1[j], b1);
          const float hi  = fminf(a2[j], b2);
          const float iou = fmaxf(hi - lo, 0.0f) / (a2[j] - a1[j] + 1e-5f);
          if (iou > 0.5f && j != best) ind[j] = 0;
        }
      }
    }

    float* sp = band ? spy : spx;
#pragma unroll
    for (int i = 0; i < NP; ++i) {
      const float f   = (float)ind[i];
      const float pz  = pv[i] * f;
      const float c1z = a1[i] * f;
      const float c2z = a2[i] * f;
      const int q1 = (int)fminf(fmaxf(floorf(c1z * dim), 0.0f), dim - 1.0f);
      const int q2 = (int)fminf(fmaxf(floorf(c2z * dim), 0.0f), dim - 1.0f);
      wsq[row * 40 + band * 20 + i]      = q1;
      wsq[row * 40 + band * 20 + 10 + i] = q2;
      sp[row * NP + i] = pz;
    }
  }
  __syncthreads();
  for (int idx = t; idx < ROWS * NP; idx += 32)
    pout[idx] = sqrtf(spx[idx] * spy[idx] + 1e-10f);
}

// ---------------------------------------------------------------------------
// Kernel 2: streaming mask/split. blockIdx.y = (b*C+c) row, each thread owns 4
// consecutive flat (w,h) elements -> b128 load of x, 2x10 b128 stores.
// The 40 per-row bounds are staged into LDS with the gfx1250 async path
// (global_load_async_to_lds_b32 + s_wait_asynccnt), then broadcast-read.
// ---------------------------------------------------------------------------
__global__ __launch_bounds__(256) void mask_split_kernel(
    const float* __restrict__ x_fft, const int* __restrict__ wsq,
    float* __restrict__ clean, float* __restrict__ dirty) {
  __shared__ int sp[40];
  const int row = blockIdx.y;
  const int tid = threadIdx.x;

  if (tid < 40) {
    const int* gp = wsq + row * 40 + tid;
    // absolute LDS byte offset of &sp[tid]
    unsigned lds_off = (unsigned)(unsigned long long)
        ((__attribute__((address_space(3))) int*)sp + tid);
    asm volatile("global_load_async_to_lds_b32 %0, %1, off"
                 :: "v"(lds_off), "v"(gp) : "memory");
  }
  asm volatile("s_wait_asynccnt 0" ::: "memory");
  __syncthreads();

  const int flat = (blockIdx.x * 256 + tid) * 4;
  if (flat >= WH) return;

  const v4f xv = *(const v4f*)(x_fft + (size_t)row * WH + flat);
  float xs[4] = {xv.x, xv.y, xv.z, xv.w};

  int w[4], h[4];
  w[0] = flat / HD;
  h[0] = flat - w[0] * HD;
#pragma unroll
  for (int e = 1; e < 4; ++e) {
    h[e] = h[e - 1] + 1;
    w[e] = w[e - 1];
    if (h[e] == HD) { h[e] = 0; w[e] += 1; }
  }

#pragma unroll
  for (int n = 0; n < NP; ++n) {
    const int x1 = sp[n],      x2 = sp[10 + n];
    const int y1 = sp[20 + n], y2 = sp[30 + n];
    v4f cv, dv;
    float c[4], d[4];
#pragma unroll
    for (int e = 0; e < 4; ++e) {
      const bool m = (w[e] >= x1 && w[e] <= x2) || (h[e] >= y1 && h[e] <= y2);
      d[e] = m ? xs[e] : 0.0f;
      c[e] = m ? 0.0f : xs[e];
    }
    cv.x = c[0]; cv.y = c[1]; cv.z = c[2]; cv.w = c[3];
    dv.x = d[0]; dv.y = d[1]; dv.z = d[2]; dv.w = d[3];
    const size_t o = (size_t)(row * NP + n) * WH + (size_t)flat;
    *(v4f*)(clean + o) = cv;
    *(v4f*)(dirty + o) = dv;
  }
}

// ---------------------------------------------------------------------------
extern "C" void kernel_launch(void* const* d_in, const int* in_sizes, int n_in,
                              void* d_out, int out_size, void* d_ws, size_t ws_size,
                              hipStream_t stream) {
  (void)in_sizes; (void)n_in; (void)out_size; (void)ws_size;
  const float* x_fft = (const float*)d_in[0];
  const float* px    = (const float*)d_in[1];
  const float* xc1   = (const float*)d_in[2];
  const float* xc2   = (const float*)d_in[3];
  const float* py    = (const float*)d_in[4];
  const float* yc1   = (const float*)d_in[5];
  const float* yc2   = (const float*)d_in[6];

  const size_t S = (size_t)ROWS * NP * WH;   // 3,962,880 elements per tensor
  float* clean = (float*)d_out;
  float* dirty = clean + S;
  float* pout  = clean + 2 * S;              // 120 elements
  int*   wsq   = (int*)d_ws;                 // 480 ints

  nms_quant_kernel<<<dim3(1), dim3(32), 0, stream>>>(px, xc1, xc2, py, yc1, yc2, wsq, pout);

  const int threads_needed = WH / 4;                       // 8256
  const int gx = (threads_needed + 255) / 256;             // 33
  mask_split_kernel<<<dim3(gx, ROWS), dim3(256), 0, stream>>>(x_fft, wsq, clean, dirty);
}